// SinkhornDistance_V2_1073741824491
// MI455X (gfx1250) — compile-verified
//
#include <hip/hip_runtime.h>
#include <hip/hip_bf16.h>

// Sinkhorn: U = exp(-C/eps) cached in ws (64MB, L2-resident on MI455X's 192MB L2).
// 100 x { nu = cons/(U mu) [wave row-matvec], mu = cons/(U^T nu) [f32 WMMA col-matvec] }
// then T = mu * U * nu^T.
// Col-matvec is K-split 4x per output tile for memory-level parallelism at L2 BW.

constexpr int NB = 16;     // batches
constexpr int NN = 1024;   // points
constexpr float EPS_INV = 20.0f;   // 1/0.05
constexpr int ITERS = 100;

typedef __attribute__((ext_vector_type(2))) float v2f;
typedef __attribute__((ext_vector_type(8))) float v8f;

// ---------------- init: U = exp(-C/eps), vectorized float4 ----------------
__global__ __launch_bounds__(256) void init_U(const float* __restrict__ C,
                                              float* __restrict__ U) {
  size_t idx = (size_t)blockIdx.x * 256 + threadIdx.x;   // float4 index
  float4 c = ((const float4*)C)[idx];
  float4 u;
  u.x = __expf(-EPS_INV * c.x);
  u.y = __expf(-EPS_INV * c.y);
  u.z = __expf(-EPS_INV * c.z);
  u.w = __expf(-EPS_INV * c.w);
  ((float4*)U)[idx] = u;
}

__global__ __launch_bounds__(256) void init_vec(float* __restrict__ v, float val) {
  v[(size_t)blockIdx.x * 256 + threadIdx.x] = val;
}

// ------------- forward: y[b,i] = cons / sum_j U[b,i,j]*v[b,j] -------------
// one wave per row; float4 lane-strided loads (coalesced); LDS-staged vector.
__global__ __launch_bounds__(256) void rowmatvec_recip(
    const float* __restrict__ U, const float* __restrict__ v,
    float* __restrict__ y, float cons) {
  __shared__ float sv[NN];
  const int tid  = threadIdx.x;
  const int wave = tid >> 5;
  const int lane = tid & 31;
  const int row  = blockIdx.x * 8 + wave;      // 0 .. NB*NN-1 (8 rows/block, same batch)
  const int b    = row >> 10;
  const int i    = row & (NN - 1);

  ((float4*)sv)[tid] = ((const float4*)(v + (size_t)b * NN))[tid];
  __syncthreads();

  const float* Ur = U + (size_t)b * NN * NN + (size_t)i * NN;
  float s = 0.0f;
#pragma unroll
  for (int c = lane * 4; c < NN; c += 128) {
    float4 u = *(const float4*)(Ur + c);
    float4 m = *(const float4*)(sv + c);
    s += u.x * m.x + u.y * m.y + u.z * m.z + u.w * m.w;
  }
  for (int off = 16; off > 0; off >>= 1) s += __shfl_xor(s, off, 32);
  if (lane == 0) y[row] = cons / s;
}

// ---------- transposed: y[b,n] = cons / sum_k U[b,k,n]*v[b,k] -------------
// V_WMMA_F32_16X16X4_F32: A[m][k] = v[k] (broadcast over rows, per-lane uniform),
// B = 4x16 tile of U read row-major (coalesced). All D rows identical; lanes
// 0-15 of acc[0] carry the 16 column results.
// Block = 256 threads = 8 waves = 2 column-tiles x 4 K-splits; partials are
// combined through LDS, then 32 threads do the reciprocal + store.
__global__ __launch_bounds__(256) void colmatvec_recip_wmma(
    const float* __restrict__ U, const float* __restrict__ v,
    float* __restrict__ y, float cons) {
  __shared__ float sv[NN];
  __shared__ float pr[8][16];
  const int tid    = threadIdx.x;
  const int wave   = tid >> 5;
  const int lane   = tid & 31;
  const int tileIB = wave >> 2;                 // 0..1: tile within block
  const int ksplit = wave & 3;                  // 0..3: K quarter
  const int tile   = blockIdx.x * 2 + tileIB;   // global 16-col tile, 0..1023
  const int b      = tile >> 6;                 // 64 tiles per batch (block never straddles)
  const int n0     = (tile & 63) << 4;

  ((float4*)sv)[tid] = ((const float4*)(v + (size_t)b * NN))[tid];
  __syncthreads();

  const float* Mb  = U + (size_t)b * NN * NN;
  // A layout (32b, 16x4): VGPR0 = K0(l0-15)/K2(l16-31), VGPR1 = K1/K3.
  const int    lo  = (lane < 16) ? 0 : 2;
  const float* bp0 = Mb + n0 + (lane & 15);
  const int    k0  = ksplit * (NN / 4);

  v8f acc = {};
#pragma unroll 4
  for (int kk = k0; kk < k0 + NN / 4; kk += 4) {
    const int kr = kk + lo;
    v2f av, bv;
    av.x = sv[kr];
    av.y = sv[kr + 1];
    const float* bp = bp0 + (size_t)kr * NN;   // rows kr, kr+1 of the 4x16 B tile
    bv.x = bp[0];
    bv.y = bp[NN];
    acc = __builtin_amdgcn_wmma_f32_16x16x4_f32(false, av, false, bv,
                                                (short)0, acc, false, false);
  }
  if (lane < 16) pr[wave][lane] = acc[0];      // partial sums for this K quarter
  __syncthreads();

  if (tid < 32) {                              // 2 tiles x 16 cols
    const int t   = tid >> 4;                  // tile within block
    const int col = tid & 15;
    float s = pr[t * 4 + 0][col] + pr[t * 4 + 1][col] +
              pr[t * 4 + 2][col] + pr[t * 4 + 3][col];
    const int tg = blockIdx.x * 2 + t;
    y[(size_t)(tg >> 6) * NN + ((tg & 63) << 4) + col] = cons / s;
  }
}

// ---------------- final: T = mu * U * nu^T, float4 vectorized -------------
__global__ __launch_bounds__(256) void final_T(
    const float* __restrict__ U, const float* __restrict__ mu,
    const float* __restrict__ nu, float* __restrict__ T) {
  size_t idx4 = (size_t)blockIdx.x * 256 + threadIdx.x;  // float4 index
  size_t flat = idx4 * 4;
  int b = (int)(flat >> 20);
  int i = (int)((flat >> 10) & (NN - 1));
  int j = (int)(flat & (NN - 1));
  float4 u  = ((const float4*)U)[idx4];
  float4 n4 = *(const float4*)(nu + (size_t)b * NN + j);
  float  m  = mu[(size_t)b * NN + i];
  float4 t;
  t.x = m * u.x * n4.x;
  t.y = m * u.y * n4.y;
  t.z = m * u.z * n4.z;
  t.w = m * u.w * n4.w;
  ((float4*)T)[idx4] = t;
}

extern "C" void kernel_launch(void* const* d_in, const int* in_sizes, int n_in,
                              void* d_out, int out_size, void* d_ws, size_t ws_size,
                              hipStream_t stream) {
  // inputs (setup_inputs order): x (B,D,N), y (B,D,N), C (B,N,N) -- only C matters
  const float* C = (const float*)d_in[2];
  float* U  = (float*)d_ws;                        // NB*NN*NN floats (64 MB)
  float* mu = U + (size_t)NB * NN * NN;            // NB*NN
  float* nu = mu + (size_t)NB * NN;                // NB*NN
  float* T  = (float*)d_out;
  const float cons = 1.0f / (float)NN;

  const int elems4 = NB * NN * NN / 4;             // 4M float4's
  init_U  <<<elems4 / 256, 256, 0, stream>>>(C, U);
  init_vec<<<NB * NN / 256, 256, 0, stream>>>(mu, cons);

  for (int it = 0; it < ITERS; ++it) {
    // nu = cons / (U @ mu)
    rowmatvec_recip     <<<NB * NN / 8,      256, 0, stream>>>(U, mu, nu, cons);
    // mu = cons / (U^T @ nu)  -- 2 tiles x 4 K-splits per block
    colmatvec_recip_wmma<<<NB * NN / 16 / 2, 256, 0, stream>>>(U, nu, mu, cons);
  }

  final_T<<<elems4 / 256, 256, 0, stream>>>(U, mu, nu, T);
}